// Orbital_Head_34479997452479
// MI455X (gfx1250) — compile-verified
//
#include <hip/hip_runtime.h>
#include <hip/hip_bf16.h>
#include <math.h>

// Problem constants (from reference)
#define BQ      8192
#define N_ELEC  16
#define EMB     256
#define N_DET   16
#define N_SPIN  8          // N_UP == N_DOWN == 8
#define DCOLS   (N_DET * N_SPIN)   // 128

typedef __attribute__((ext_vector_type(2))) float v2f;
typedef __attribute__((ext_vector_type(8))) float v8f;

#define A_STRIDE 260   // 16x256 tile rows padded: conflict-free b64 frag reads
#define P_STRIDE 132   // 16x128 phi stage rows padded

// One wave (32 lanes) handles 16 rows = 2 batches x 8 electrons.
// Block = 4 waves = 8 batches. Grid = BQ/8 = 1024 blocks per spin side.
__global__ __launch_bounds__(128) void ferminet_orbital_head(
    const float* __restrict__ h,        // (B, 16, 256)
    const int*   __restrict__ sidx,     // (8,)
    const float* __restrict__ r_ae,     // (B, 8)
    const float* __restrict__ pi_v,     // (128,)
    const float* __restrict__ sigma_v,  // (128,)
    const float* __restrict__ W,        // (128, 256)
    const float* __restrict__ bias_v,   // (128,)
    float*       __restrict__ out)      // (B,)
{
    __shared__ float sW[2][16 * A_STRIDE];         // double-buffered W column tile
    __shared__ float sA[4][16 * A_STRIDE];         // per-wave gathered A tile
    __shared__ float sP[4][16 * P_STRIDE];         // per-wave phi staging (16 x 128)

    const int tid  = threadIdx.x;
    const int lane = tid & 31;
    const int w    = tid >> 5;                     // wave id 0..3
    const int b0   = blockIdx.x * 8 + w * 2;       // first batch of this wave

    // ---- Stage gathered A tile: rows r=0..15 -> (b0 + r/8, electron sidx[r%8]) ----
    for (int t = lane; t < 16 * 64; t += 32) {     // 1024 float4 chunks
        const int r  = t >> 6;
        const int c4 = t & 63;
        const int b  = b0 + (r >> 3);
        const int el = sidx[r & 7];
        const float4 v = *(const float4*)(h + ((size_t)(b * N_ELEC + el) * EMB) + c4 * 4);
        *(float4*)(&sA[w][r * A_STRIDE + c4 * 4]) = v;
    }

    // ---- Prime W double buffer with tile 0 (block-cooperative) ----
    {
        for (int i = 0; i < 8; ++i) {
            const int t  = tid + i * 128;
            const int r  = t >> 6;
            const int c4 = t & 63;
            const float4 v = *(const float4*)(W + ((size_t)r * EMB) + c4 * 4);
            *(float4*)(&sW[0][r * A_STRIDE + c4 * 4]) = v;
        }
    }
    __syncthreads();                               // A tiles + sW[0] visible

    const int nloc  = lane & 15;                   // col within 16-wide N tile / row M%16
    const int half  = lane >> 4;                   // 0: lanes 0-15, 1: lanes 16-31
    const int fragk = 2 * half;                    // K sub-offset of this lane's fragment

    // ---- Loop over 8 column tiles of the 128 orbital outputs ----
    for (int nt = 0; nt < 8; ++nt) {
        const int cur = nt & 1;

        // Issue next tile's global loads NOW; they fly during the WMMA phase.
        float4 wpre[8];
        if (nt < 7) {
#pragma unroll
            for (int i = 0; i < 8; ++i) {
                const int t  = tid + i * 128;
                const int r  = t >> 6;
                const int c4 = t & 63;
                wpre[i] = *(const float4*)(W + ((size_t)((nt + 1) * 16 + r) * EMB) + c4 * 4);
            }
        }

        v8f acc = {0.f, 0.f, 0.f, 0.f, 0.f, 0.f, 0.f, 0.f};
        const float* aBase = &sA[w][nloc * A_STRIDE + fragk];
        const float* bBase = &sW[cur][nloc * A_STRIDE + fragk];
#pragma unroll
        for (int kk = 0; kk < 64; ++kk) {
            const int kb = kk * 4;
            v2f afrag = *(const v2f*)(aBase + kb);  // A[M=nloc, K=kb+fragk .. +1]
            v2f bfrag = *(const v2f*)(bBase + kb);  // B[K=kb+fragk.., N=nloc] = W[ncol, k]
            acc = __builtin_amdgcn_wmma_f32_16x16x4_f32(
                false, afrag, false, bfrag, (short)0, acc, false, false);
        }

        // ---- Epilogue: bias + envelope, deposit into phi staging LDS ----
        // D layout: element e -> row M = e + 8*half, col N = nloc.
        // Row M -> batch b0+half (fixed per lane), electron n = e.
        const int   d   = nt * 16 + nloc;
        const float sg  = sigma_v[d];
        const float pp  = pi_v[d];
        const float bb  = bias_v[d];
        const float* ra = r_ae + (size_t)(b0 + half) * N_SPIN;
#pragma unroll
        for (int e = 0; e < 8; ++e) {
            const float val = (acc[e] + bb) * __expf(-ra[e] * sg) * pp;
            sP[w][(e + 8 * half) * P_STRIDE + d] = val;
        }

        // Deposit prefetched W tile into the other buffer, then one barrier.
        if (nt < 7) {
#pragma unroll
            for (int i = 0; i < 8; ++i) {
                const int t  = tid + i * 128;
                const int r  = t >> 6;
                const int c4 = t & 63;
                *(float4*)(&sW[cur ^ 1][r * A_STRIDE + c4 * 4]) = wpre[i];
            }
        }
        __syncthreads();   // next buffer ready AND everyone done reading sW[cur]
    }

    // ---- Per-lane 8x8 LU with partial pivoting, in LDS ----
    // lane -> (local batch lb = lane>>4, determinant det = lane&15)
    const int lb  = half;
    const int det = nloc;
    float* M = &sP[w][(size_t)(lb * 8) * P_STRIDE + det * 8];
#define AT(n, i) M[(n) * P_STRIDE + (i)]
    float logabs = 0.f;
    for (int k = 0; k < 8; ++k) {
        int p = k;
        float mv = fabsf(AT(k, k));
        for (int r = k + 1; r < 8; ++r) {
            const float v = fabsf(AT(r, k));
            if (v > mv) { mv = v; p = r; }
        }
        if (p != k) {
            for (int c = k; c < 8; ++c) {
                const float t = AT(k, c); AT(k, c) = AT(p, c); AT(p, c) = t;
            }
        }
        const float piv = AT(k, k);
        logabs += __logf(fabsf(piv));
        const float inv = 1.0f / piv;
        for (int r = k + 1; r < 8; ++r) {
            const float f = AT(r, k) * inv;
            for (int c = k + 1; c < 8; ++c) AT(r, c) -= f * AT(k, c);
        }
    }
#undef AT

    // ---- logsumexp over the 16 determinants (16-lane subgroup reduction) ----
    float mx = logabs;
#pragma unroll
    for (int off = 8; off >= 1; off >>= 1)
        mx = fmaxf(mx, __shfl_xor(mx, off, 16));
    float s = __expf(logabs - mx);
#pragma unroll
    for (int off = 8; off >= 1; off >>= 1)
        s += __shfl_xor(s, off, 16);
    const float res = isinf(mx) ? mx : mx + __logf(s);

    if (nloc == 0) out[b0 + lb] = res;
}

extern "C" void kernel_launch(void* const* d_in, const int* in_sizes, int n_in,
                              void* d_out, int out_size, void* d_ws, size_t ws_size,
                              hipStream_t stream) {
    (void)in_sizes; (void)n_in; (void)out_size; (void)d_ws; (void)ws_size;
    const float* h        = (const float*)d_in[0];
    const int*   up_idx   = (const int*)  d_in[1];
    const int*   dn_idx   = (const int*)  d_in[2];
    const float* r_up     = (const float*)d_in[3];
    const float* r_dn     = (const float*)d_in[4];
    const float* pi_up    = (const float*)d_in[5];
    const float* sg_up    = (const float*)d_in[6];
    const float* pi_dn    = (const float*)d_in[7];
    const float* sg_dn    = (const float*)d_in[8];
    const float* W_up     = (const float*)d_in[9];
    const float* b_up     = (const float*)d_in[10];
    const float* W_dn     = (const float*)d_in[11];
    const float* b_dn     = (const float*)d_in[12];
    float* out = (float*)d_out;

    dim3 grid(BQ / 8), block(128);
    ferminet_orbital_head<<<grid, block, 0, stream>>>(
        h, up_idx, r_up, pi_up, sg_up, W_up, b_up, out);          // up -> out[0..B)
    ferminet_orbital_head<<<grid, block, 0, stream>>>(
        h, dn_idx, r_dn, pi_dn, sg_dn, W_dn, b_dn, out + BQ);     // down -> out[B..2B)
}